// LigthGCNLayer_66305705115855
// MI455X (gfx1250) — compile-verified
//
#include <hip/hip_runtime.h>
#include <hip/hip_bf16.h>

#define N_NODES 100000
#define N_EDGES 3200000
#define D_FEAT  128

#define BLOCK 256                 // 8 waves (wave32)
#define TILE  1024                // edges staged per block
#define EDGES_PER_WAVE (TILE / (BLOCK / 32))   // 128
#define PF_DIST 4                 // prefetch distance (edges)

typedef unsigned int u32x4 __attribute__((ext_vector_type(4)));
typedef int          i32x4 __attribute__((ext_vector_type(4)));
typedef int          i32x8 __attribute__((ext_vector_type(8)));

// ---- LDS address helper: 32-bit LDS byte offset ----
typedef __attribute__((address_space(3))) void lds_void;
__device__ __forceinline__ unsigned lds_off(void* p) {
  return (unsigned)(unsigned long long)(lds_void*)p;
}

// ---- guaranteed hardware f32 atomic add (no-return form, STOREcnt-tracked) ----
__device__ __forceinline__ void atomic_add_f32(float* p, float v) {
  asm volatile("global_atomic_add_f32 %0, %1, off"
               :: "v"(p), "v"(v)
               : "memory");
}

// ---- 1-D TDM copy: n_tile elems into LDS; elems beyond n_tensor read as 0 ----
// D# built per CDNA5 ISA §8 (group0: count/lds_addr/global_addr/type,
// group1: data_size/tensor_dim0/tile_dim0, 1-D so all other dims/strides 0).
// This toolchain's builtin is the 6-arg amdgpu-toolchain form:
//   (u32x4 g0, i32x8 g1, i32x4 g2, i32x4 g3, i32x8 ext, i32 cpol)
__device__ __forceinline__ void tdm_load_1d(void* lds_dst, const void* gsrc,
                                            unsigned n_tile, unsigned n_tensor) {
  unsigned long long ga = (unsigned long long)gsrc;
  u32x4 g0;
  g0.x = 0x1u;                                   // count=1, user D#, no gather
  g0.y = lds_off(lds_dst);                       // lds_addr      [63:32]
  g0.z = (unsigned)ga;                           // global_addr   [95:64]
  g0.w = (unsigned)((ga >> 32) & 0x01FFFFFFu)    // global_addr   [120:96]
       | (2u << 30);                             // type=2        [127:126]
  i32x8 g1;
  g1[0] = (int)(2u << 16);                       // data_size=4B  [17:16]; wg_mask=0
  g1[1] = (int)((n_tensor & 0xFFFFu) << 16);     // tensor_dim0 lo [63:48]
  g1[2] = (int)((n_tensor >> 16) & 0xFFFFu);     // tensor_dim0 hi [79:64]; tensor_dim1 lo=0
  g1[3] = (int)((n_tile & 0xFFFFu) << 16);       // tile_dim0     [127:112]
  g1[4] = 0;                                     // tile_dim1=0, tile_dim2=0 (1-D)
  g1[5] = 0;                                     // tensor_dim0_stride (unused, 1-D)
  g1[6] = 0;
  g1[7] = 0;
  i32x4 gz  = {0, 0, 0, 0};                      // groups 2/3: dims 2..4 unused
  i32x8 gz8 = {0, 0, 0, 0, 0, 0, 0, 0};          // uncharacterized extended operand
  __builtin_amdgcn_tensor_load_to_lds(g0, g1, gz, gz, gz8, 0);
}

// ------------------------- COO scatter-add SpMM -------------------------
// out[rows[e]] += vals[e] * embeds[cols[e]]   (E=3.2M, D=128). Wave32: lane l
// owns features {l, l+32, l+64, l+96} so every gather load and every atomic
// instruction touches exactly one full 128B cache line.
__global__ __launch_bounds__(BLOCK) void coo_spmm_kernel(
    const int*   __restrict__ rows,
    const int*   __restrict__ cols,
    const float* __restrict__ vals,
    const float* __restrict__ embeds,
    float*       __restrict__ out) {

  __shared__ int   s_rows[TILE];
  __shared__ int   s_cols[TILE];
  __shared__ float s_vals[TILE];

  const int tid  = threadIdx.x;
  const int lane = tid & 31;
  const int wave = tid >> 5;
  const int base = blockIdx.x * TILE;    // first edge of this block's tile

  // ---- Stage the edge tile with the Tensor Data Mover: one DMA per array.
  // tensor_dim0 = remaining edges, so OOB elements of the tail tile read as 0.
  if (wave == 0) {
    unsigned remain = (unsigned)(N_EDGES - base);
    tdm_load_1d(s_rows, rows + base, TILE, remain);
    tdm_load_1d(s_cols, cols + base, TILE, remain);
    tdm_load_1d(s_vals, vals + base, TILE, remain);
    __builtin_amdgcn_s_wait_tensorcnt(0);
  }
  __syncthreads();                       // publish LDS tile to all 8 waves

  // ---- Each wave owns a contiguous slice of the tile ----
  const int wbase = wave * EDGES_PER_WAVE;

  for (int j = 0; j < EDGES_PER_WAVE; ++j) {
    int e = base + wbase + j;
    if (e >= N_EDGES) break;             // uniform across the wave

    int   col = s_cols[wbase + j];       // uniform LDS broadcasts
    int   row = s_rows[wbase + j];
    float v   = s_vals[wbase + j];

    // Prefetch a future edge's embed row; lane l prefetches byte offset 16*l,
    // so the 32 lanes cover all 4 cache lines of the 512B row.
    int jp = j + PF_DIST;
    if (jp < EDGES_PER_WAVE && (base + wbase + jp) < N_EDGES) {
      int colp = s_cols[wbase + jp];
      __builtin_prefetch(embeds + (size_t)colp * D_FEAT + lane * 4, 0, 0);
    }

    // Gather: 4 loads, each one full 128B line across the wave
    const float* g = embeds + (size_t)col * D_FEAT + lane;
    float g0 = g[0];
    float g1 = g[32];
    float g2 = g[64];
    float g3 = g[96];

    // Scatter: 4 hardware f32 atomics, each one full 128B line across the wave
    float* o = out + (size_t)row * D_FEAT + lane;
    atomic_add_f32(o +  0, v * g0);
    atomic_add_f32(o + 32, v * g1);
    atomic_add_f32(o + 64, v * g2);
    atomic_add_f32(o + 96, v * g3);
  }
}

// ------------------------- zero the output (poisoned by harness) -------------------------
__global__ __launch_bounds__(BLOCK) void zero_out_kernel(float4* __restrict__ out4, int n4) {
  int i = blockIdx.x * BLOCK + threadIdx.x;
  if (i < n4) out4[i] = make_float4(0.f, 0.f, 0.f, 0.f);
}

extern "C" void kernel_launch(void* const* d_in, const int* in_sizes, int n_in,
                              void* d_out, int out_size, void* d_ws, size_t ws_size,
                              hipStream_t stream) {
  const int*   rows   = (const int*)  d_in[0];
  const int*   cols   = (const int*)  d_in[1];
  const float* vals   = (const float*)d_in[2];
  const float* embeds = (const float*)d_in[3];
  float*       out    = (float*)d_out;

  const int n4 = (N_NODES * D_FEAT) / 4;
  zero_out_kernel<<<(n4 + BLOCK - 1) / BLOCK, BLOCK, 0, stream>>>((float4*)out, n4);

  const int nblocks = (N_EDGES + TILE - 1) / TILE;
  coo_spmm_kernel<<<nblocks, BLOCK, 0, stream>>>(rows, cols, vals, embeds, out);
}